// SampleSubset_24137716204259
// MI455X (gfx1250) — compile-verified
//
#include <hip/hip_runtime.h>

// Relaxed top-k subset sampling (ermongroup/subsets), B=256 rows of N=65536.
// One workgroup per row, 512 threads (16 wave32 waves -> 4 waves/SIMD ->
// 256-VGPR budget). w[] lives entirely in VGPRs (128 regs); khot lives in
// LDS (256 KB of the WGP's 320 KB), thread-private, lane-interleaved
// (conflict-free ds_*_b128). Per-group data-dependence anchors stop the
// scheduler from hoisting all 128 exp ops at once. Final output flush uses
// CDNA5 GLOBAL_STORE_ASYNC_FROM_LDS_B128 (ASYNCcnt) to DMA LDS -> HBM.

#define K_SELECT 10
#define TAU0     0.1f
#define EPS_TINY 1.17549435082228751e-38f   // np.finfo(float32).tiny
#define LN2      0.69314718055994531f
#define LOG2E    1.44269504088896341f
#define NEG_FLT_MAX (-3.402823466e+38f)

constexpr int N_ELEM = 65536;
constexpr int TPB    = 512;               // 16 wave32 waves
constexpr int V4     = N_ELEM / 4;        // 16384 float4 per row
constexpr int PER    = V4 / TPB;          // 32 float4 per thread (128 floats)
constexpr int KGQ    = 8;                 // chunks per LDS base (off <= 57344 B)
constexpr int GRP    = 4;                 // chunks per anchored group
constexpr int NWAVE  = TPB / 32;          // 16 waves per block

__device__ __forceinline__ void sched_fence() {
  asm volatile("" ::: "memory");          // memory-op scheduling barrier
}

__device__ __forceinline__ float ln_fast(float x) {
  // v_log_f32 computes log2; scale to natural log
  return __builtin_amdgcn_logf(x) * LN2;
}
__device__ __forceinline__ float exp2_fast(float x) {
  return __builtin_amdgcn_exp2f(x);       // v_exp_f32
}

__device__ __forceinline__ float wave_max(float v) {
  #pragma unroll
  for (int o = 16; o > 0; o >>= 1) v = fmaxf(v, __shfl_xor(v, o, 32));
  return v;
}
__device__ __forceinline__ float wave_sum(float v) {
  #pragma unroll
  for (int o = 16; o > 0; o >>= 1) v += __shfl_xor(v, o, 32);
  return v;
}

// Block-wide reductions (16 wave partials; lanes >= NWAVE padded with the
// identity). Callers alternate between the two scratch arrays so the other
// reduction's barriers provide WAR protection on red[NWAVE].
__device__ __forceinline__ float block_max(float v, float* red, int lane, int wave) {
  v = wave_max(v);
  if (lane == 0) red[wave] = v;
  __syncthreads();
  if (wave == 0) {
    float x = (lane < NWAVE) ? red[lane] : NEG_FLT_MAX;
    x = wave_max(x);
    if (lane == 0) red[NWAVE] = x;
  }
  __syncthreads();
  return red[NWAVE];
}
__device__ __forceinline__ float block_sum(float v, float* red, int lane, int wave) {
  v = wave_sum(v);
  if (lane == 0) red[wave] = v;
  __syncthreads();
  if (wave == 0) {
    float x = (lane < NWAVE) ? red[lane] : 0.f;
    x = wave_sum(x);
    if (lane == 0) red[NWAVE] = x;
  }
  __syncthreads();
  return red[NWAVE];
}

__global__ __launch_bounds__(TPB, 1)
void SampleSubset_24137716204259_kernel(const float* __restrict__ logits,
                                        const float* __restrict__ noise,
                                        float* __restrict__ out)
{
  __shared__ float4 khl[PER * TPB];         // 256 KB: khot accumulator in LDS
  __shared__ float  redMax[NWAVE + 1];
  __shared__ float  redSum[NWAVE + 1];

  const int t    = threadIdx.x;
  const int lane = t & 31;
  const int wave = t >> 5;
  const size_t rowBase = (size_t)blockIdx.x * (size_t)N_ELEM;

  const float4* __restrict__ lg = reinterpret_cast<const float4*>(logits + rowBase);
  const float4* __restrict__ un = reinterpret_cast<const float4*>(noise  + rowBase);
  float4*       __restrict__ ot = reinterpret_cast<float4*>(out + rowBase);

  // 4 LDS base pointers; each covers 8 chunks via immediate offsets
  // (q*TPB float4 = q*8192 B <= 57344 B, fits the 16-bit DS offset field).
  float4* kb[PER / KGQ];
  #pragma unroll
  for (int g = 0; g < PER / KGQ; ++g) kb[g] = &khl[g * KGQ * TPB + t];

  float4 w[PER];     // perturbed logits, updated each iteration (128 VGPRs)

  // ---- init: w = logits + (-log(-log(clip(u)))), zero khot, track max ----
  // Fenced groups of 4 chunks (8 in-flight b128 loads) bound peak pressure.
  float lmax = NEG_FLT_MAX;
  #pragma unroll
  for (int g = 0; g < PER / GRP; ++g) {
    float4 l[GRP], u[GRP];
    #pragma unroll
    for (int q = 0; q < GRP; ++q) {
      const int j = g * GRP + q;
      l[q] = lg[t + j * TPB];
      u[q] = un[t + j * TPB];
    }
    sched_fence();
    // anchor: this group's logs depend on lmax so far (exact: 0*finite = +/-0)
    const float z = 0.0f * lmax;
    #pragma unroll
    for (int q = 0; q < GRP; ++q) {
      const int j = g * GRP + q;
      float4 x;
      x.x = l[q].x - ln_fast(z - ln_fast(fmaxf(u[q].x, EPS_TINY)));
      x.y = l[q].y - ln_fast(z - ln_fast(fmaxf(u[q].y, EPS_TINY)));
      x.z = l[q].z - ln_fast(z - ln_fast(fmaxf(u[q].z, EPS_TINY)));
      x.w = l[q].w - ln_fast(z - ln_fast(fmaxf(u[q].w, EPS_TINY)));
      w[j] = x;
      lmax = fmaxf(lmax, fmaxf(fmaxf(x.x, x.y), fmaxf(x.z, x.w)));
      kb[j / KGQ][(j % KGQ) * TPB] = make_float4(0.f, 0.f, 0.f, 0.f);
    }
    sched_fence();
  }

  float m = block_max(lmax, redMax, lane, wave);
  const float sc = LOG2E / TAU0;   // softmax(w/tau) via exp2

  #pragma unroll 1
  for (int it = 0; it < K_SELECT; ++it) {
    const float ms = m * sc;

    // ---- pass 1: row sum of exp((w - m)/tau) ----
    // Anchor each group's exps on the running sum so at most one group of
    // exp results is live at a time (0*lsum == +0 exactly; lsum >= 0).
    float lsum = 0.f;
    #pragma unroll
    for (int g = 0; g < PER / GRP; ++g) {
      const float msg = ms + 0.0f * lsum;
      float s0 = 0.f, s1 = 0.f, s2 = 0.f, s3 = 0.f;
      #pragma unroll
      for (int q = 0; q < GRP; ++q) {
        const int j = g * GRP + q;
        s0 += exp2_fast(w[j].x * sc - msg);
        s1 += exp2_fast(w[j].y * sc - msg);
        s2 += exp2_fast(w[j].z * sc - msg);
        s3 += exp2_fast(w[j].w * sc - msg);
      }
      lsum += (s0 + s1) + (s2 + s3);
    }
    const float rsum = block_sum(lsum, redSum, lane, wave);
    const float rinv = 1.0f / rsum;

    // ---- pass 2: onehot, accumulate khot (LDS), mask w, track next max ----
    float nmax = NEG_FLT_MAX;
    #pragma unroll
    for (int g = 0; g < PER / GRP; ++g) {
      // anchor: this group's exps depend on nmax so far (1 + (+/-0) == 1)
      const float rg = rinv * (1.0f + 0.0f * nmax);
      #pragma unroll
      for (int q = 0; q < GRP; ++q) {
        const int j = g * GRP + q;
        float4 x = w[j];
        float4 oh;
        oh.x = exp2_fast(x.x * sc - ms) * rg;
        oh.y = exp2_fast(x.y * sc - ms) * rg;
        oh.z = exp2_fast(x.z * sc - ms) * rg;
        oh.w = exp2_fast(x.w * sc - ms) * rg;
        float4* kp = kb[j / KGQ] + (j % KGQ) * TPB;   // thread-private slot
        float4 k = *kp;
        k.x += oh.x; k.y += oh.y; k.z += oh.z; k.w += oh.w;
        *kp = k;
        x.x += ln_fast(fmaxf(1.f - oh.x, EPS_TINY));
        x.y += ln_fast(fmaxf(1.f - oh.y, EPS_TINY));
        x.z += ln_fast(fmaxf(1.f - oh.z, EPS_TINY));
        x.w += ln_fast(fmaxf(1.f - oh.w, EPS_TINY));
        w[j] = x;
        nmax = fmaxf(nmax, fmaxf(fmaxf(x.x, x.y), fmaxf(x.z, x.w)));
      }
      sched_fence();    // bound concurrent ds rmw live ranges
    }
    m = block_max(nmax, redMax, lane, wave);
  }

  // ---- async DMA flush: LDS khot -> global out [B, N, 1] ----
  // khl's lane-interleaved layout is byte-identical to the row's linear
  // layout: LDS byte (j*TPB+t)*16 <-> global byte (j*TPB+t)*16. GVS mode
  // adds the instruction offset to BOTH addresses (ISA 10.x async ops),
  // so one VADDR (t*16) + one LDS-address VGPR + per-chunk imm offsets.
  {
    // ISA 10.2: generic LDS pointer low 32 bits == LDS byte address.
    const unsigned int vlds  = (unsigned int)(unsigned long long)(const void*)khl
                             + (unsigned int)t * 16u;
    const unsigned int vaddr = (unsigned int)t * 16u;
    asm volatile("s_wait_dscnt 0x0" ::: "memory");   // LDS writes visible to DMA
    #pragma unroll
    for (int j = 0; j < PER; ++j) {
      asm volatile("global_store_async_from_lds_b128 %0, %1, %2 offset:%3"
                   :
                   : "v"(vaddr), "v"(vlds), "s"(ot), "i"(j * TPB * 16)
                   : "memory");
    }
    asm volatile("s_wait_asynccnt 0x0" ::: "memory");
  }
}

extern "C" void kernel_launch(void* const* d_in, const int* in_sizes, int n_in,
                              void* d_out, int out_size, void* d_ws, size_t ws_size,
                              hipStream_t stream) {
  const float* logits = (const float*)d_in[0];   // [B, N, 1] fp32
  const float* noise  = (const float*)d_in[1];   // [B, N]    fp32
  float* out = (float*)d_out;                    // [B, N, 1] fp32
  const int B = in_sizes[1] / N_ELEM;            // 256
  SampleSubset_24137716204259_kernel<<<B, TPB, 0, stream>>>(logits, noise, out);
}